// AdaptivePooling_57741540327799
// MI455X (gfx1250) — compile-verified
//
#include <hip/hip_runtime.h>
#include <cstdint>

// Adaptive avg pool 225x225 -> 7x7, NHWC (32, 225, 225, 256) f32.
// Memory-bound (~1.66 GB in, ~72 us floor at 23.3 TB/s). Strategy:
// gfx1250 async global->LDS B128 staging (ASYNCcnt-gated, double buffered,
// per-wave-private so the hot loop has no barriers), float4 accumulation,
// LDS tree reduction across row-groups.

namespace {
constexpr int IN_H = 225;
constexpr int IN_W = 225;
constexpr int CH   = 256;
constexpr int OUT  = 7;
constexpr int POS  = 7;                                // async positions per batch per wave
constexpr int WAVES = 8;
constexpr int SLOT_BYTES = 512;                        // 128 channels * 4B (one wave, one (h,w))
constexpr int STAGE_PER_WAVE = 2 * POS * SLOT_BYTES;   // 7168 B (double buffered)
}

__global__ __launch_bounds__(256, 1)
void AdaptivePooling_57741540327799_kernel(const float* __restrict__ x,
                                           float* __restrict__ out) {
  __shared__ __align__(16) char stage[WAVES * STAGE_PER_WAVE];  // 56 KB staging
  __shared__ float4 red[WAVES][32];                             // 4 KB reduction

  const int cell = blockIdx.x;                 // (b*7 + oh)*7 + ow
  const int ow = cell % OUT;
  const int oh = (cell / OUT) % OUT;
  const int b  = cell / (OUT * OUT);

  // Bin edges: start = floor(i*225/7), end = ceil((i+1)*225/7)
  const int h0 = (oh * IN_H) / OUT;
  const int h1 = ((oh + 1) * IN_H + OUT - 1) / OUT;
  const int w0 = (ow * IN_W) / OUT;
  const int w1 = ((ow + 1) * IN_W + OUT - 1) / OUT;
  const int bh = h1 - h0;                      // 33 for all bins here
  const int bw = w1 - w0;

  const int tid  = threadIdx.x;
  const int lane = tid & 31;
  const int wave = tid >> 5;
  const int half = wave & 1;                   // channel half: 0 -> c[0,128), 1 -> c[128,256)
  const int rg   = wave >> 1;                  // row group: rows h0+rg, h0+rg+4, ...

  const int nrows = (bh > rg) ? ((bh - rg + 3) >> 2) : 0;
  const int n  = nrows * bw;                   // real positions for this wave
  const int nb = (n + POS - 1) / POS;          // batches (padded to POS)

  // LDS byte offset of this wave's staging region (low 32 bits of the generic
  // LDS address are the within-allocation offset on gfx1250).
  const uint32_t stage_base =
      (uint32_t)(uintptr_t)(&stage[0]) + (uint32_t)(wave * STAGE_PER_WAVE);
  // Per-lane global base: 4 consecutive channels, 512B contiguous per wave.
  const float* gbase = x + ((size_t)b * IN_H * IN_W) * (size_t)CH
                         + (size_t)(half * 128 + lane * 4);

  int ir = 0, jc = 0;  // issue cursor: row-in-group, column

  auto issue_batch = [&](int k) {
#pragma unroll
    for (int j = 0; j < POS; ++j) {
      const bool live = (ir < nrows);
      const int iru = live ? ir : 0;           // clamp padded issues to a safe addr
      const int jcu = live ? jc : 0;
      const int hh = h0 + rg + (iru << 2);
      const int ww = w0 + jcu;
      const float* gp = gbase + ((size_t)hh * IN_W + ww) * (size_t)CH;
      const uint32_t loff =
          stage_base + (uint32_t)((((k & 1) * POS + j) << 9) + (lane << 4));
      asm volatile("global_load_async_to_lds_b128 %0, %1, off"
                   :
                   : "v"(loff), "v"(gp)
                   : "memory");
      if (live) {
        if (++jc == bw) { jc = 0; ++ir; }
      }
    }
  };

  float4 acc;
  acc.x = 0.f; acc.y = 0.f; acc.z = 0.f; acc.w = 0.f;

  auto consume_batch = [&](int k) {
#pragma unroll
    for (int j = 0; j < POS; ++j) {
      const int p = k * POS + j;               // wave-uniform predicate
      if (p < n) {
        const float4 v = *(const float4*)(
            &stage[(size_t)wave * STAGE_PER_WAVE +
                   (size_t)((((k & 1) * POS + j) << 9) + (lane << 4))]);
        acc.x += v.x; acc.y += v.y; acc.z += v.z; acc.w += v.w;
      }
    }
  };

  // Software-pipelined: issue batch k+1, wait for batch k (in-order
  // completion => ASYNCcnt <= POS means the previous batch landed), consume.
  issue_batch(0);
  for (int k = 0; k + 1 < nb; ++k) {
    issue_batch(k + 1);
    asm volatile("s_wait_asynccnt 7" ::: "memory");
    consume_batch(k);
  }
  asm volatile("s_wait_asynccnt 0" ::: "memory");
  consume_batch(nb - 1);

  // Cross row-group reduction in LDS.
  red[wave][lane] = acc;
  __syncthreads();

  if (wave < 2) {  // rg == 0; half == wave
    const float4 a0 = red[wave][lane];
    const float4 a1 = red[wave + 2][lane];
    const float4 a2 = red[wave + 4][lane];
    const float4 a3 = red[wave + 6][lane];
    const float inv = 1.0f / (float)(bh * bw);
    float4 o;
    o.x = (a0.x + a1.x + a2.x + a3.x) * inv;
    o.y = (a0.y + a1.y + a2.y + a3.y) * inv;
    o.z = (a0.z + a1.z + a2.z + a3.z) * inv;
    o.w = (a0.w + a1.w + a2.w + a3.w) * inv;
    float* op = out + (size_t)cell * CH + (size_t)(wave * 128 + lane * 4);
    *(float4*)op = o;
  }
}

extern "C" void kernel_launch(void* const* d_in, const int* in_sizes, int n_in,
                              void* d_out, int out_size, void* d_ws, size_t ws_size,
                              hipStream_t stream) {
  const float* x = (const float*)d_in[0];
  float* out = (float*)d_out;
  const int blocks = 32 * OUT * OUT;  // one block per output cell
  hipLaunchKernelGGL(AdaptivePooling_57741540327799_kernel,
                     dim3(blocks), dim3(256), 0, stream, x, out);
}